// GraphConv_12721693131105
// MI455X (gfx1250) — compile-verified
//
#include <hip/hip_runtime.h>

// Degree-bucketed GraphConv for MI455X (gfx1250, wave32, WMMA).
// For deg d>=1: out = [rel_sum | self] @ [W[2(d-1)] ; W[2(d-1)+1]]  (K=256 fused GEMM,
// exploiting that the two W blocks are contiguous in memory) + biases.
// deg 0: out = self @ W[12] + b[12]  (K=128).
// FP32 path via V_WMMA_F32_16X16X4_F32 (A 16x4, B 4x16, C/D 16x16 f32).

#define N_PER_DEG 30000
#define MAX_DEG 6
#define NFEAT 128
#define ROWS_PER_BLOCK 48                                // 3 M-tiles of 16
#define BLOCKS_PER_BUCKET (N_PER_DEG / ROWS_PER_BLOCK)   // 625, exact
#define LDS_STRIDE 260   // 256 data + 4 pad: row r banks start at (4r)%64 -> conflict-free

typedef __attribute__((ext_vector_type(2))) float v2f;
typedef __attribute__((ext_vector_type(4))) float v4f;
typedef __attribute__((ext_vector_type(8))) float v8f;

struct AdjPtrs { const int* a[6]; };

__global__ __launch_bounds__(128) void graphconv_wmma_f32(
    const float* __restrict__ atoms,   // (210000, 128)
    const float* __restrict__ W,       // (13, 128, 128)
    const float* __restrict__ bias,    // (13, 128)
    AdjPtrs adj,                       // adj1..adj6: (30000, d) int32
    float* __restrict__ out)           // (210000, 128)
{
    __shared__ float A_lds[ROWS_PER_BLOCK * LDS_STRIDE];   // ~48.8 KB

    const int blk    = blockIdx.x;
    const int bucket = blk / BLOCKS_PER_BUCKET;          // 0..6 == degree
    const int rb     = blk % BLOCKS_PER_BUCKET;
    const int row0   = rb * ROWS_PER_BLOCK;              // row within bucket
    const long grow0 = (long)bucket * N_PER_DEG + row0;  // global row

    const int tid  = threadIdx.x;
    const int r16  = tid >> 3;   // 0..15
    const int seg  = tid & 7;    // 16-feature segment

    // LDS row layout: bucket>0 -> [rel(0..127) | self(128..255)];  bucket 0 -> self at 0..127
    const int selfOff = (bucket > 0) ? NFEAT : 0;

    const int* __restrict__ Aj = nullptr;
    if      (bucket == 1) Aj = adj.a[0];
    else if (bucket == 2) Aj = adj.a[1];
    else if (bucket == 3) Aj = adj.a[2];
    else if (bucket == 4) Aj = adj.a[3];
    else if (bucket == 5) Aj = adj.a[4];
    else if (bucket == 6) Aj = adj.a[5];

    // ---- Phase 1: stage [rel | self] tile rows into LDS ----
    #pragma unroll
    for (int i = 0; i < 3; ++i) {
        const int row = r16 + 16 * i;                    // 0..47

        // self segment
        {
            const v4f* __restrict__ src =
                (const v4f*)(atoms + (grow0 + row) * (long)NFEAT) + seg * 4;
            float* dst = &A_lds[row * LDS_STRIDE + selfOff + seg * 16];
            v4f t0 = src[0], t1 = src[1], t2 = src[2], t3 = src[3];
            *(v4f*)(dst + 0)  = t0;
            *(v4f*)(dst + 4)  = t1;
            *(v4f*)(dst + 8)  = t2;
            *(v4f*)(dst + 12) = t3;
        }

        // gather-sum rel segment (deg >= 1)
        if (bucket > 0) {
            const int deg = bucket;
            v4f s0 = {0.f,0.f,0.f,0.f}, s1 = {0.f,0.f,0.f,0.f};
            v4f s2 = {0.f,0.f,0.f,0.f}, s3 = {0.f,0.f,0.f,0.f};
            const long abase = (long)(row0 + row) * deg;
            for (int j = 0; j < deg; ++j) {
                const int nb = Aj[abase + j];
                const v4f* __restrict__ src =
                    (const v4f*)(atoms + (long)nb * NFEAT) + seg * 4;
                s0 += src[0]; s1 += src[1]; s2 += src[2]; s3 += src[3];
            }
            float* dst = &A_lds[row * LDS_STRIDE + seg * 16];
            *(v4f*)(dst + 0)  = s0;
            *(v4f*)(dst + 4)  = s1;
            *(v4f*)(dst + 8)  = s2;
            *(v4f*)(dst + 12) = s3;
        }
    }

    __syncthreads();

    // ---- Phase 2: fused-K WMMA f32 16x16x4 ----
    const int lane  = tid & 31;
    const int wave  = tid >> 5;          // 0..3 -> N-tiles n0, n0+16
    const int col   = lane & 15;
    const int khalf = (lane >> 4) << 1;  // lanes 16..31 take K+2,K+3
    const int n0    = wave * 32;

    // B base: deg buckets use the contiguous [W_rel ; W_self] 256x128 block.
    const int ibase = (bucket > 0) ? (2 * (bucket - 1)) : (2 * MAX_DEG);
    const int kmax  = (bucket > 0) ? (2 * NFEAT) : NFEAT;

    v8f acc00 = {}, acc01 = {};   // M-tile 0, N-tiles n0 / n0+16
    v8f acc10 = {}, acc11 = {};   // M-tile 1
    v8f acc20 = {}, acc21 = {};   // M-tile 2

    {
        const float* __restrict__ w0 =
            W + (long)ibase * NFEAT * NFEAT + (long)khalf * NFEAT + n0 + col;
        const float* a0 = &A_lds[(col +  0) * LDS_STRIDE + khalf];
        const float* a1 = &A_lds[(col + 16) * LDS_STRIDE + khalf];
        const float* a2 = &A_lds[(col + 32) * LDS_STRIDE + khalf];

        #pragma unroll 4
        for (int k = 0; k < kmax; k += 4) {
            v2f fa0 = *(const v2f*)(a0 + k);             // ds_load_b64, conflict-free
            v2f fa1 = *(const v2f*)(a1 + k);
            v2f fa2 = *(const v2f*)(a2 + k);
            v2f b0, b1;
            b0.x = w0[(long)k * NFEAT];                  // B row k+khalf
            b0.y = w0[(long)(k + 1) * NFEAT];            // B row k+khalf+1
            b1.x = w0[(long)k * NFEAT + 16];
            b1.y = w0[(long)(k + 1) * NFEAT + 16];

            acc00 = __builtin_amdgcn_wmma_f32_16x16x4_f32(
                false, fa0, false, b0, (short)0, acc00, false, false);
            acc01 = __builtin_amdgcn_wmma_f32_16x16x4_f32(
                false, fa0, false, b1, (short)0, acc01, false, false);
            acc10 = __builtin_amdgcn_wmma_f32_16x16x4_f32(
                false, fa1, false, b0, (short)0, acc10, false, false);
            acc11 = __builtin_amdgcn_wmma_f32_16x16x4_f32(
                false, fa1, false, b1, (short)0, acc11, false, false);
            acc20 = __builtin_amdgcn_wmma_f32_16x16x4_f32(
                false, fa2, false, b0, (short)0, acc20, false, false);
            acc21 = __builtin_amdgcn_wmma_f32_16x16x4_f32(
                false, fa2, false, b1, (short)0, acc21, false, false);
        }
    }

    // ---- Phase 3: bias + store ----
    // Biases: deg buckets add b[2(d-1)] + b[2(d-1)+1]; deg0 adds b[12].
    float bv = bias[(long)ibase * NFEAT + n0 + col];
    if (bucket > 0) bv += bias[(long)(ibase + 1) * NFEAT + n0 + col];

    const int mhi = (lane >> 4) * 8;     // C/D: vgpr j -> M = j + 8*(lane>=16), N = lane&15
    float* __restrict__ ob = out + grow0 * NFEAT + n0 + col;
    #pragma unroll
    for (int j = 0; j < 8; ++j) {
        const long m = j + mhi;
        ob[(m +  0) * NFEAT]      = acc00[j] + bv;
        ob[(m +  0) * NFEAT + 16] = acc01[j] + bv;
        ob[(m + 16) * NFEAT]      = acc10[j] + bv;
        ob[(m + 16) * NFEAT + 16] = acc11[j] + bv;
        ob[(m + 32) * NFEAT]      = acc20[j] + bv;
        ob[(m + 32) * NFEAT + 16] = acc21[j] + bv;
    }
}

extern "C" void kernel_launch(void* const* d_in, const int* in_sizes, int n_in,
                              void* d_out, int out_size, void* d_ws, size_t ws_size,
                              hipStream_t stream) {
    (void)in_sizes; (void)n_in; (void)out_size; (void)d_ws; (void)ws_size;
    const float* atoms = (const float*)d_in[0];
    const float* W     = (const float*)d_in[1];
    const float* bias  = (const float*)d_in[2];
    // d_in[3] = deg_slice: structural constant (start = d*30000, count = 30000)
    AdjPtrs adj;
    adj.a[0] = (const int*)d_in[4];
    adj.a[1] = (const int*)d_in[5];
    adj.a[2] = (const int*)d_in[6];
    adj.a[3] = (const int*)d_in[7];
    adj.a[4] = (const int*)d_in[8];
    adj.a[5] = (const int*)d_in[9];

    dim3 grid((MAX_DEG + 1) * BLOCKS_PER_BUCKET);  // 7 * 625 = 4375
    graphconv_wmma_f32<<<grid, 128, 0, stream>>>(
        atoms, W, bias, adj, (float*)d_out);
}